// dMaSIFConv_1898375545077
// MI455X (gfx1250) — compile-verified
//
#include <hip/hip_runtime.h>
#include <math.h>

// ---------------------------------------------------------------------------
// dMaSIFConv for MI455X (gfx1250, wave32, WMMA).
//   T1 = leaky(features @ W_in1^T + b_in1)        [wmma f32 16x16x4 GEMM]
//   T2 = leaky(T1 @ W_in2^T + b_in2)              [wmma GEMM]
//   F  = GroupNorm(T2)
//   P0,P1 = pairwise quasi-geodesic conv(F)       [wmma, dominant: 22 GFLOP]
//   G1 = leaky((P0+P1) @ W_out1^T + b_out1)       [wmma GEMM, folds j-split sum]
//   G2 = leaky(G1 @ W_out2^T + b_out2)            [wmma GEMM]
//   out= GroupNorm(G2)
//
// fp32 WMMA layouts (CDNA5 ISA 7.12.2):
//   A 16x4 : lane L, m=L%16, vgpr v holds K = v + 2*(L>=16)
//   B 4x16 : lane L, n=L%16, vgpr v holds K = v + 2*(L>=16)
//   C/D    : vgpr r, lanes 0-15 -> M=r, lanes 16-31 -> M=r+8, N=lane%16
// Pairwise trick: w>0 so w*relu(A2 h + c2) = relu(A2'(w h;w)), K=9 pad 12.
// Channel permutation c = 4*(lane%16) + ct makes the per-j f-row load one b128.
// ---------------------------------------------------------------------------

typedef float v2f __attribute__((ext_vector_type(2)));
typedef float v4f __attribute__((ext_vector_type(4)));
typedef float v8f __attribute__((ext_vector_type(8)));

#define NPTS  4096
#define HCH   64

// ---------------------------------------------------------------------------
// Y = leaky(X @ W^T + b). grid 256 (16-row tiles), block 128 (4 col tiles).
// Two independent WMMA chains break the 16-deep accumulator latency chain.
// ---------------------------------------------------------------------------
__global__ void __launch_bounds__(128) gemm_leaky_64(
    const float* __restrict__ X, const float* __restrict__ W,
    const float* __restrict__ bias, float* __restrict__ Y) {
  const int lane = threadIdx.x & 31;
  const int ct   = threadIdx.x >> 5;
  const int hb   = lane >> 4;
  const int li   = lane & 15;
  const int rbase = blockIdx.x * 16;
  const int cb    = ct * 16;

  const float* Xrow = X + (rbase + li) * 64 + 2 * hb;
  const float* Wrow = W + (cb    + li) * 64 + 2 * hb;

  v8f acc0 = {}, acc1 = {};
#pragma unroll
  for (int k0 = 0; k0 < 64; k0 += 8) {
    v2f a0 = *(const v2f*)(Xrow + k0);
    v2f b0 = *(const v2f*)(Wrow + k0);
    acc0 = __builtin_amdgcn_wmma_f32_16x16x4_f32(false, a0, false, b0,
                                                 (short)0, acc0, false, false);
    v2f a1 = *(const v2f*)(Xrow + k0 + 4);
    v2f b1 = *(const v2f*)(Wrow + k0 + 4);
    acc1 = __builtin_amdgcn_wmma_f32_16x16x4_f32(false, a1, false, b1,
                                                 (short)0, acc1, false, false);
  }

  const float bv  = bias[cb + li];
  const int  row0 = rbase + hb * 8;
#pragma unroll
  for (int r = 0; r < 8; ++r) {
    float y = acc0[r] + acc1[r] + bv;
    y = fmaxf(y, 0.2f * y);
    Y[(row0 + r) * 64 + cb + li] = y;
  }
}

// Same GEMM but A = X0 + X1 on the fly (merges the split-j partial sums).
__global__ void __launch_bounds__(128) gemm_leaky_64_sum2(
    const float* __restrict__ X0, const float* __restrict__ X1,
    const float* __restrict__ W, const float* __restrict__ bias,
    float* __restrict__ Y) {
  const int lane = threadIdx.x & 31;
  const int ct   = threadIdx.x >> 5;
  const int hb   = lane >> 4;
  const int li   = lane & 15;
  const int rbase = blockIdx.x * 16;
  const int cb    = ct * 16;

  const float* X0row = X0 + (rbase + li) * 64 + 2 * hb;
  const float* X1row = X1 + (rbase + li) * 64 + 2 * hb;
  const float* Wrow  = W  + (cb    + li) * 64 + 2 * hb;

  v8f acc0 = {}, acc1 = {};
#pragma unroll
  for (int k0 = 0; k0 < 64; k0 += 8) {
    v2f a0 = *(const v2f*)(X0row + k0) + *(const v2f*)(X1row + k0);
    v2f b0 = *(const v2f*)(Wrow + k0);
    acc0 = __builtin_amdgcn_wmma_f32_16x16x4_f32(false, a0, false, b0,
                                                 (short)0, acc0, false, false);
    v2f a1 = *(const v2f*)(X0row + k0 + 4) + *(const v2f*)(X1row + k0 + 4);
    v2f b1 = *(const v2f*)(Wrow + k0 + 4);
    acc1 = __builtin_amdgcn_wmma_f32_16x16x4_f32(false, a1, false, b1,
                                                 (short)0, acc1, false, false);
  }

  const float bv  = bias[cb + li];
  const int  row0 = rbase + hb * 8;
#pragma unroll
  for (int r = 0; r < 8; ++r) {
    float y = acc0[r] + acc1[r] + bv;
    y = fmaxf(y, 0.2f * y);
    Y[(row0 + r) * 64 + cb + li] = y;
  }
}

// ---------------------------------------------------------------------------
// GroupNorm over (N, C): stats per group over (N x C/G). grid 4, block 256.
// ---------------------------------------------------------------------------
__global__ void __launch_bounds__(256) groupnorm64(
    const float* __restrict__ X, const float* __restrict__ gw,
    const float* __restrict__ gb, float* __restrict__ Y) {
  const int g = blockIdx.x;
  const int t = threadIdx.x;
  __shared__ float s_sum[256], s_sq[256];
  __shared__ float s_mean, s_rstd;

  float sum = 0.f, sq = 0.f;
  for (int e = t; e < NPTS * 16; e += 256) {
    const int n = e >> 4, c = (g << 4) + (e & 15);
    const float x = X[n * 64 + c];
    sum += x; sq += x * x;
  }
  s_sum[t] = sum; s_sq[t] = sq;
  __syncthreads();
  for (int off = 128; off > 0; off >>= 1) {
    if (t < off) { s_sum[t] += s_sum[t + off]; s_sq[t] += s_sq[t + off]; }
    __syncthreads();
  }
  if (t == 0) {
    const float m   = s_sum[0] * (1.0f / 65536.0f);
    const float var = s_sq[0] * (1.0f / 65536.0f) - m * m;
    s_mean = m; s_rstd = rsqrtf(var + 1e-5f);
  }
  __syncthreads();
  const float m = s_mean, rs = s_rstd;
  for (int e = t; e < NPTS * 16; e += 256) {
    const int n = e >> 4, c = (g << 4) + (e & 15);
    Y[n * 64 + c] = (X[n * 64 + c] - m) * rs * gw[c] + gb[c];
  }
}

// ---------------------------------------------------------------------------
// Pairwise interaction. grid 512 = 256 i-tiles x 2 j-halves; block = 8 waves.
// Per j: 12 x v_wmma_f32_16x16x4_f32; epilogue acc += relu(D)*f (D-layout).
// Channel map: tile ct, lane li -> channel c = 4*li + ct (f loads = 1 b128).
// ---------------------------------------------------------------------------
__global__ void __launch_bounds__(256) pairwise_conv(
    const float* __restrict__ pts, const float* __restrict__ nuv,
    const float* __restrict__ F, const float* __restrict__ A1,
    const float* __restrict__ c1, const float* __restrict__ A2,
    const float* __restrict__ c2, float* __restrict__ P0,
    float* __restrict__ P1) {
  const int lane  = threadIdx.x & 31;
  const int wv    = threadIdx.x >> 5;         // wave 0..7
  const int hb    = lane >> 4;
  const int li    = lane & 15;
  const int kb    = hb * 2;
  const int itile = blockIdx.x & 255;
  const int jhalf = blockIdx.x >> 8;          // 0 or 1
  float* __restrict__ P = jhalf ? P1 : P0;
  const int i0 = itile * 16;
  const int i  = i0 + li;
  const float sc = 0.70710678118654752f;      // 1/(sqrt(2)*RADIUS)

  // --- per-i state ---
  const float pix = pts[i * 3 + 0] * sc;
  const float piy = pts[i * 3 + 1] * sc;
  const float piz = pts[i * 3 + 2] * sc;
  const float* nv = nuv + i * 9;
  const float u0 = nv[0], u1 = nv[1], u2 = nv[2];   // n_i
  const float v0 = nv[3], v1 = nv[4], v2 = nv[5];
  const float q0 = nv[6], q1 = nv[7], q2 = nv[8];

  // A1 rows / c1 for this lane's K-slots: kb, kb+1, kb+4, kb+5
  float a1v[4][3], c1v[4];
#pragma unroll
  for (int qq = 0; qq < 4; ++qq) {
    const int kk = kb + (qq & 1) + (qq >> 1) * 4;
    a1v[qq][0] = A1[kk * 3 + 0];
    a1v[qq][1] = A1[kk * 3 + 1];
    a1v[qq][2] = A1[kk * 3 + 2];
    c1v[qq]    = c1[kk];
  }

  // B operands from A2' = [A2 | c2 | 0...]; permuted channel c = 4*li + ct.
  v2f Bv[4][3];
#pragma unroll
  for (int ct = 0; ct < 4; ++ct) {
    const int c = 4 * li + ct;
    const float* a2r = A2 + c * 8;
    v2f b0; b0[0] = a2r[kb];     b0[1] = a2r[kb + 1];
    v2f b1; b1[0] = a2r[kb + 4]; b1[1] = a2r[kb + 5];
    v2f b2; b2[0] = hb ? 0.0f : c2[c]; b2[1] = 0.0f;
    Bv[ct][0] = b0; Bv[ct][1] = b1; Bv[ct][2] = b2;
  }

  v8f acc[4]; const v8f zero8 = {};
#pragma unroll
  for (int ct = 0; ct < 4; ++ct) acc[ct] = zero8;

  const int jbeg = jhalf * 2048 + wv * 256, jend = jbeg + 256;
  for (int j = jbeg; j < jend; ++j) {
    // j is wave-uniform: force scalar (SMEM) loads for broadcast geometry.
    const int ju = __builtin_amdgcn_readfirstlane(j);
    const float* pj  = pts + ju * 3;
    const float* njv = nuv + ju * 9;
    const float pjx = pj[0] * sc, pjy = pj[1] * sc, pjz = pj[2] * sc;
    const float nj0 = njv[0], nj1 = njv[1], nj2 = njv[2];

    const float dx = pjx - pix, dy = pjy - piy, dz = pjz - piz;
    const float dot = u0 * nj0 + u1 * nj1 + u2 * nj2;
    const float r2  = dx * dx + dy * dy + dz * dz;
    const float tt  = 2.0f - dot;
    const float w   = __expf(-(r2 * tt * tt));     // gaussian window, w > 0

    const float X0 = u0 * dx + u1 * dy + u2 * dz;  // local frame coords
    const float X1 = v0 * dx + v1 * dy + v2 * dz;
    const float X2 = q0 * dx + q1 * dy + q2 * dz;

    float h[4];
#pragma unroll
    for (int qq = 0; qq < 4; ++qq)                 // w*relu(A1.X + c1)
      h[qq] = fmaxf(0.0f, c1v[qq] + a1v[qq][0] * X0 + a1v[qq][1] * X1 +
                          a1v[qq][2] * X2) * w;

    v2f Aop0; Aop0[0] = h[0]; Aop0[1] = h[1];
    v2f Aop1; Aop1[0] = h[2]; Aop1[1] = h[3];
    v2f Aop2; Aop2[0] = hb ? 0.0f : w; Aop2[1] = 0.0f;

    // one contiguous 16B f load per lane (channels 4*li .. 4*li+3)
    const float* Fj = F + ju * 64 + 4 * li;
    v4f fv = *(const v4f*)Fj;
    __builtin_prefetch(Fj + 8 * 64, 0, 3);         // global_prefetch, ~8 j ahead

#pragma unroll
    for (int ct = 0; ct < 4; ++ct) {
      v8f d = __builtin_amdgcn_wmma_f32_16x16x4_f32(false, Aop0, false,
                  Bv[ct][0], (short)0, zero8, false, false);
      d = __builtin_amdgcn_wmma_f32_16x16x4_f32(false, Aop1, false,
                  Bv[ct][1], (short)0, d, false, false);
      d = __builtin_amdgcn_wmma_f32_16x16x4_f32(false, Aop2, false,
                  Bv[ct][2], (short)0, d, false, false);
#pragma unroll
      for (int r = 0; r < 8; ++r)                  // acc += relu(w*Xc)*f_j[c]
        acc[ct][r] = fmaf(fmaxf(d[r], 0.0f), fv[ct], acc[ct][r]);
    }
  }

  // --- cross-wave reduction via LDS ---
  __shared__ float lds[8192];                      // 8 wv * 4 ct * 32 lane * 8 r
  float* my = lds + wv * 1024 + lane * 8;
#pragma unroll
  for (int r = 0; r < 8; ++r) {
    my[r]       = acc[0][r];
    my[256 + r] = acc[1][r];
    my[512 + r] = acc[2][r];
    my[768 + r] = acc[3][r];
  }
  __syncthreads();

  const int t    = threadIdx.x;
  const int rsel = (t & 1) * 4;
  const int lsel = (t >> 1) & 31;
  const int csel = t >> 6;
  float o0 = 0.f, o1 = 0.f, o2 = 0.f, o3 = 0.f;
#pragma unroll
  for (int wq = 0; wq < 8; ++wq) {
    const float* p = lds + wq * 1024 + csel * 256 + lsel * 8 + rsel;
    o0 += p[0]; o1 += p[1]; o2 += p[2]; o3 += p[3];
  }
  const int mh = (lsel >> 4) * 8;                  // D-layout decode
  const int c  = 4 * (lsel & 15) + csel;           // permuted channel map
  const int ib = i0 + rsel + mh;
  P[(ib + 0) * 64 + c] = o0;
  P[(ib + 1) * 64 + c] = o1;
  P[(ib + 2) * 64 + c] = o2;
  P[(ib + 3) * 64 + c] = o3;
}

// ---------------------------------------------------------------------------
extern "C" void kernel_launch(void* const* d_in, const int* in_sizes, int n_in,
                              void* d_out, int out_size, void* d_ws,
                              size_t ws_size, hipStream_t stream) {
  (void)in_sizes; (void)n_in; (void)out_size; (void)ws_size;
  const float* points   = (const float*)d_in[0];
  const float* nuv      = (const float*)d_in[1];
  const float* feats    = (const float*)d_in[2];
  const float* W_in1    = (const float*)d_in[3];
  const float* b_in1    = (const float*)d_in[4];
  const float* W_in2    = (const float*)d_in[5];
  const float* b_in2    = (const float*)d_in[6];
  const float* gn_in_w  = (const float*)d_in[7];
  const float* gn_in_b  = (const float*)d_in[8];
  const float* A1       = (const float*)d_in[9];
  const float* c1       = (const float*)d_in[10];
  const float* A2       = (const float*)d_in[11];
  const float* c2       = (const float*)d_in[12];
  const float* W_out1   = (const float*)d_in[13];
  const float* b_out1   = (const float*)d_in[14];
  const float* W_out2   = (const float*)d_in[15];
  const float* b_out2   = (const float*)d_in[16];
  const float* gn_out_w = (const float*)d_in[17];
  const float* gn_out_b = (const float*)d_in[18];
  float* out = (float*)d_out;

  float* T1 = (float*)d_ws;                 // each buffer: 4096*64 floats
  float* T2 = T1 + NPTS * HCH;
  float* Fb = T2 + NPTS * HCH;
  float* Pa = Fb + NPTS * HCH;
  float* Pb = Pa + NPTS * HCH;

  gemm_leaky_64<<<256, 128, 0, stream>>>(feats, W_in1, b_in1, T1);
  gemm_leaky_64<<<256, 128, 0, stream>>>(T1, W_in2, b_in2, T2);
  groupnorm64 <<<4, 256, 0, stream>>>(T2, gn_in_w, gn_in_b, Fb);
  pairwise_conv<<<512, 256, 0, stream>>>(points, nuv, Fb, A1, c1, A2, c2,
                                         Pa, Pb);
  gemm_leaky_64_sum2<<<256, 128, 0, stream>>>(Pa, Pb, W_out1, b_out1, T1);
  gemm_leaky_64<<<256, 128, 0, stream>>>(T1, W_out2, b_out2, T2);
  groupnorm64 <<<4, 256, 0, stream>>>(T2, gn_out_w, gn_out_b, out);
}